// ScorePredictor_13511967113601
// MI455X (gfx1250) — compile-verified
//
#include <hip/hip_runtime.h>

typedef __attribute__((ext_vector_type(2))) float v2f;
typedef __attribute__((ext_vector_type(4))) float v4f;
typedef __attribute__((ext_vector_type(8))) float v8f;

// One wave32 processes 16 edges:
//   lane L (0..15)  handles edge m=L, holds cols [it*8+0 .. it*8+3] of its rows
//   lane L+16       handles edge m=L, holds cols [it*8+4 .. it*8+7]
// This matches the 32-bit A-matrix 16x4 WMMA layout (lanes L / L+16 share row M=L),
// so b128 gathers feed V_WMMA_F32_16X16X4_F32 directly with B = ones (reduction).
__global__ __launch_bounds__(256) void edge_dot_wmma_kernel(
    const float* __restrict__ h,
    const long long* __restrict__ src,
    const long long* __restrict__ dst,
    float* __restrict__ out,
    int E)
{
    const int lane          = threadIdx.x & 31;
    const int waveInBlock   = threadIdx.x >> 5;
    const int wavesPerBlock = blockDim.x >> 5;
    const int numGroups     = (E + 15) >> 4;
    const int groupStride   = gridDim.x * wavesPerBlock;

    const int half = lane >> 4;   // 0: cols 0-3 of chunk, 1: cols 4-7
    const int m    = lane & 15;   // edge-in-group / matrix row

    for (int g = blockIdx.x * wavesPerBlock + waveInBlock; g < numGroups; g += groupStride) {
        const int  e     = g * 16 + m;
        const int  ec    = (e < E) ? e : 0;          // clamp for tail (EXEC must stay full for WMMA)
        const long long srow = src[ec];
        const long long drow = dst[ec];

        const char* sbase = (const char*)h + (srow << 9) + (half << 4); // row*512 + half*16
        const char* dbase = (const char*)h + (drow << 9) + (half << 4);

        v8f acc = {};
        const v2f ones = {1.0f, 1.0f};

        #pragma unroll
        for (int it = 0; it < 16; ++it) {           // 16 chunks x 8 cols = D=128
            v4f a = *(const v4f*)(sbase + it * 32); // global_load_b128, 16B/lane, coalesced gather
            v4f b = *(const v4f*)(dbase + it * 32);
            v4f p = a * b;                          // elementwise product (layout-invariant)

            v2f p0 = {p.x, p.y};
            v2f p1 = {p.z, p.w};
            // D = A * ones + C  ->  every column of D accumulates sum_k A[m][k]
            acc = __builtin_amdgcn_wmma_f32_16x16x4_f32(
                false, p0, false, ones, (short)0, acc, false, false);
            acc = __builtin_amdgcn_wmma_f32_16x16x4_f32(
                false, p1, false, ones, (short)0, acc, false, false);
        }

        // C/D 16x16 f32 layout: VGPR j: lanes 0-15 -> M=j, N=lane; lanes 16-31 -> M=8+j.
        // Column N=0 is held by lane 0 (edges 0..7) and lane 16 (edges 8..15).
        if ((lane & 15) == 0) {
            const int base = g * 16 + half * 8;
            #pragma unroll
            for (int j = 0; j < 8; ++j) {
                const int eo = base + j;
                if (eo < E) out[eo] = acc[j];
            }
        }
    }
}

extern "C" void kernel_launch(void* const* d_in, const int* in_sizes, int n_in,
                              void* d_out, int out_size, void* d_ws, size_t ws_size,
                              hipStream_t stream) {
    const float*     h   = (const float*)d_in[0];
    const long long* src = (const long long*)d_in[1];
    const long long* dst = (const long long*)d_in[2];
    float*           out = (float*)d_out;

    const int E = in_sizes[1];                 // number of edges
    const int numGroups = (E + 15) / 16;       // 16 edges per wave
    const int wavesPerBlock = 8;               // 256 threads = 8 wave32
    int blocks = (numGroups + wavesPerBlock - 1) / wavesPerBlock;
    if (blocks < 1) blocks = 1;

    edge_dot_wmma_kernel<<<blocks, 256, 0, stream>>>(h, src, dst, out, E);
}